// OrientedBank2D_78125455114588
// MI455X (gfx1250) — compile-verified
//
#include <hip/hip_runtime.h>

// Submanifold 3x3 one-hot conv == 8-neighbor gather on a sparse 4096x4096 lattice.
// Dense value grid (0 = inactive) with 1-cell zero halo in d_ws:
//   1) clear padded grid (b128 RT stores -> grid resident in 192MB L2)
//   2) scatter feats[i] -> grid[(r+1, c+1)]  (coords RT-cached for reuse by 3)
//   3) gather 8 halo-safe neighbors per point (6 VMEM loads), NT b128 stores out
// No bounds checks, no valid masks, no index LUT. Streaming (read-once /
// write-once) traffic uses gfx1250 non-temporal TH hints so the 67MB grid and
// the 16MB coords array own the L2.

typedef int   v2i __attribute__((ext_vector_type(2)));
typedef float v4f __attribute__((ext_vector_type(4)));
typedef float v2f __attribute__((ext_vector_type(2), aligned(4)));  // allow 4B-aligned pair loads

static constexpr int       GRID_N = 4096;
static constexpr int       PADW   = GRID_N + 8;   // 4104: halo + rows stay 16B aligned
static constexpr int       PADH   = GRID_N + 2;   // one halo row top & bottom
static constexpr long long NCELLS = (long long)PADW * (long long)PADH;  // ~16.8M floats (~67.3 MB)
static constexpr int       NQUAD  = (int)(NCELLS / 4);

__global__ void __launch_bounds__(256) sm_clear(v4f* __restrict__ g, int n4) {
    int i = blockIdx.x * blockDim.x + threadIdx.x;
    if (i < n4) {
        v4f z = {0.0f, 0.0f, 0.0f, 0.0f};
        g[i] = z;   // RT store: we WANT the cleared grid cached in L2
    }
}

__global__ void __launch_bounds__(256) sm_scatter(const v2i* __restrict__ coords,
                                                  const float* __restrict__ feats,
                                                  float* __restrict__ grid, int n) {
    int i = blockIdx.x * blockDim.x + threadIdx.x;
    if (i >= n) return;
    v2i   rc = coords[i];                               // RT: keep in L2, gather re-reads it
    float f  = __builtin_nontemporal_load(&feats[i]);   // read-once: th:NT
    grid[(rc.x + 1) * PADW + (rc.y + 1)] = f;           // RT store -> L2 resident
}

__global__ void __launch_bounds__(256) sm_gather(const v2i* __restrict__ coords,
                                                 const float* __restrict__ grid,
                                                 float* __restrict__ out, int n) {
    int i = blockIdx.x * blockDim.x + threadIdx.x;
    if (i >= n) return;
    v2i rc = coords[i];                                 // L2 hit (cached by sm_scatter)
    const float* __restrict__ p = grid + (rc.x + 1) * PADW + (rc.y + 1);
    // 6 loads cover all 8 neighbors:
    //   top row   (r+1): pair at c-1 -> {(1,-1),(1,0)}, scalar at c+1 -> (1,1)
    //   bottom row(r-1): pair at c-1 -> {(-1,-1),(-1,0)}, scalar at c+1 -> (-1,1)
    //   middle row     : scalars at c-1, c+1
    v2f   tp = *(const v2f*)(p + PADW - 1);
    float tr = p[ PADW + 1];
    v2f   bp = *(const v2f*)(p - PADW - 1);
    float br = p[-PADW + 1];
    float ml = p[-1];
    float mr = p[ 1];
    // DIRS8 order: (1,0) (-1,0) (0,1) (0,-1) (1,1) (1,-1) (-1,1) (-1,-1)
    v4f a = {tp.y, bp.y, mr, ml};
    v4f b = {tr,   tp.x, br, bp.x};
    v4f* __restrict__ o = (v4f*)(out + (size_t)i * 8);  // 16B-aligned (i*32 bytes)
    __builtin_nontemporal_store(a, o);                  // write-once stream: th:NT
    __builtin_nontemporal_store(b, o + 1);
}

extern "C" void kernel_launch(void* const* d_in, const int* in_sizes, int n_in,
                              void* d_out, int out_size, void* d_ws, size_t ws_size,
                              hipStream_t stream) {
    const v2i*   coords = (const v2i*)d_in[0];     // (N,2) int32
    const float* feats  = (const float*)d_in[1];   // (N,1) float32
    float*       out    = (float*)d_out;           // (N,8) float32
    float*       grid   = (float*)d_ws;            // padded dense grid, ~67.3 MB
    const int    n      = in_sizes[0] / 2;

    const int threads = 256;
    const int cblocks = (NQUAD + threads - 1) / threads;
    const int pblocks = (n + threads - 1) / threads;

    sm_clear  <<<cblocks, threads, 0, stream>>>((v4f*)grid, NQUAD);
    sm_scatter<<<pblocks, threads, 0, stream>>>(coords, feats, grid, n);
    sm_gather <<<pblocks, threads, 0, stream>>>(coords, grid, out, n);
}